// AttnDecoderCNN_58411555226249
// MI455X (gfx1250) — compile-verified
//
#include <hip/hip_runtime.h>
#include <hip/hip_bf16.h>
#include <math.h>

typedef __attribute__((ext_vector_type(2))) float v2f;
typedef __attribute__((ext_vector_type(4))) float v4f;
typedef __attribute__((ext_vector_type(8))) float v8f;

#define BB 128
#define EE 512
#define HH 512
#define SS 2048
#define VV 83
#define KK 3
#define PADIDX 82

// pe3[e, pos]: even e -> sin(pos * f), odd e -> cos(pos * f), f = 10000^(-(e&~1)/E)
__device__ __forceinline__ float pe_val(int e, int pos) {
    float f = __expf(-9.210340371976184f * (float)(e & ~1) * (1.0f / (float)EE));
    float arg = (float)pos * f;
    return (e & 1) ? __cosf(arg) : __sinf(arg);
}

// const_y[o] = sum_e sum_{k=0,1} (emb[PAD,e] + pe(e,k)) * conv_w[o,e,k]
__global__ void const_y_kernel(const float* __restrict__ emb,
                               const float* __restrict__ conv_w,
                               float* __restrict__ const_y) {
    int o = blockIdx.x * 256 + threadIdx.x;     // 1024 total
    const float* w  = conv_w + (size_t)o * EE * KK;
    const float* ep = emb + (size_t)PADIDX * EE;
    float acc = 0.0f;
    for (int e = 0; e < EE; ++e) {
        float base = ep[e];
        acc += (base + pe_val(e, 0)) * w[e * KK + 0];
        acc += (base + pe_val(e, 1)) * w[e * KK + 1];
    }
    const_y[o] = acc;
}

// x2[b,e] = emb[idx[b], e] + pe(e, 2)   (layer-0 last column, pe-added)
__global__ void init_x2_kernel(const float* __restrict__ emb,
                               const int* __restrict__ idx,
                               float* __restrict__ x2) {
    int i = blockIdx.x * 256 + threadIdx.x;     // B*E
    int b = i >> 9, e = i & (EE - 1);
    x2[i] = emb[(size_t)idx[b] * EE + e] + pe_val(e, 2);
}

// Y[128,1024] = X2[128,512] * W2^T + const_y + conv_b   via V_WMMA_F32_16X16X4_F32
// one wave per 16x16 tile; grid (M/16=8, N/16=64), block 32.
__global__ void conv_wmma_kernel(const float* __restrict__ x2,      // [B,E]
                                 const float* __restrict__ conv_w,  // [2H,E,K]
                                 const float* __restrict__ const_y, // [2H]
                                 const float* __restrict__ conv_b,  // [2H]
                                 float* __restrict__ y)             // [B,2H]
{
    const int m0   = blockIdx.x * 16;
    const int n0   = blockIdx.y * 16;
    const int lane = threadIdx.x;
    const int g    = lane >> 4;       // half-wave select
    const int ln   = lane & 15;

    v8f acc = {};
    // A 16x4 frag: lanes 0-15 & 16-31 both hold M=ln; VGPR0:K={0|2}, VGPR1:K={1|3}
    // -> per lane, (K=2g, K=2g+1) is a contiguous, 8B-aligned float2
    const v2f* Arow = (const v2f*)(x2 + (size_t)(m0 + ln) * EE);
    // B 4x16 frag mirrored: w2[n][e] = conv_w[(n*E + e)*K + 2]  (stride-3 scalars)
    const float* Wrow = conv_w + ((size_t)(n0 + ln) * EE) * KK + 2;

    for (int k0 = 0; k0 < EE; k0 += 4) {
        int ka = k0 + 2 * g;
        v2f a = Arow[ka >> 1];
        v2f b;
        b.x = Wrow[(size_t)ka * KK];
        b.y = Wrow[(size_t)(ka + 1) * KK];
        acc = __builtin_amdgcn_wmma_f32_16x16x4_f32(
            false, a, false, b, (short)0, acc, false, false);
    }

    int n = n0 + ln;
    float bias = const_y[n] + conv_b[n];
#pragma unroll
    for (int r = 0; r < 8; ++r) {
        int m = m0 + r + 8 * g;       // C/D layout: VGPR r -> M=r (lanes<16), M=r+8
        y[(size_t)m * (2 * HH) + n] = acc[r] + bias;
    }
}

// glu[b,h] = y[b,h]*sigmoid(y[b,h+H]); d[b,h] = glu + x2[b,h]
__global__ void glu_d_kernel(const float* __restrict__ y,
                             const float* __restrict__ x2,
                             float* __restrict__ glu,
                             float* __restrict__ dvec) {
    int i = blockIdx.x * 256 + threadIdx.x;     // B*H
    int b = i >> 9, h = i & (HH - 1);
    float a  = y[(size_t)b * (2 * HH) + h];
    float bb = y[(size_t)b * (2 * HH) + HH + h];
    float gv = a * (1.0f / (1.0f + __expf(-bb)));
    glu[i]  = gv;
    dvec[i] = gv + x2[i];
}

// scores[b,s] = sum_h d[b,h] * enc_out[b,h,s]; block: one b, 512 s values (v4f/thread).
// Encoder stream is single-use (536 MB >> 192 MB L2): non-temporal loads.
__global__ void scores_kernel(const float* __restrict__ dvec,
                              const float* __restrict__ enc_out,
                              float* __restrict__ scores) {
    __shared__ float dl[HH];
    const int b = blockIdx.x;
    const int t = threadIdx.x;                  // 128 threads
    for (int i = t; i < HH; i += 128) dl[i] = dvec[(size_t)b * HH + i];
    __syncthreads();
    const int s0 = blockIdx.y * 512 + t * 4;
    const v4f* base = (const v4f*)(enc_out + (size_t)b * HH * SS + s0);
    v4f acc = {0.f, 0.f, 0.f, 0.f};
    for (int h = 0; h < HH; ++h) {
        v4f v = __builtin_nontemporal_load(&base[(size_t)h * (SS / 4)]);
        float w = dl[h];
        acc.x += w * v.x; acc.y += w * v.y; acc.z += w * v.z; acc.w += w * v.w;
    }
    *(v4f*)(scores + (size_t)b * SS + s0) = acc;
}

// in-place softmax over S per row
__global__ void softmax_kernel(float* __restrict__ sc) {
    const int b = blockIdx.x;
    const int t = threadIdx.x;                  // 256
    __shared__ float red[256];
    float* row = sc + (size_t)b * SS;
    float m = -INFINITY;
    for (int i = t; i < SS; i += 256) m = fmaxf(m, row[i]);
    red[t] = m; __syncthreads();
    for (int o = 128; o > 0; o >>= 1) { if (t < o) red[t] = fmaxf(red[t], red[t + o]); __syncthreads(); }
    m = red[0]; __syncthreads();
    float s = 0.f;
    for (int i = t; i < SS; i += 256) { float e = __expf(row[i] - m); row[i] = e; s += e; }
    red[t] = s; __syncthreads();
    for (int o = 128; o > 0; o >>= 1) { if (t < o) red[t] += red[t + o]; __syncthreads(); }
    float inv = 1.0f / red[0];
    for (int i = t; i < SS; i += 256) row[i] *= inv;
}

// c[b,h] = sum_s attn[b,s]*enc_tot[b,h,s]; h = glu + c; x2_next = h + pe(h,2)
// enc_tot is single-use streaming (NT); attn is reused 512x per b (keep RT).
__global__ void context_kernel(const float* __restrict__ attn,
                               const float* __restrict__ enc_tot,
                               const float* __restrict__ glu,
                               float* __restrict__ hout,
                               float* __restrict__ x2next) {
    const int b    = blockIdx.x;
    const int h    = blockIdx.y * 8 + (threadIdx.x >> 5);   // one wave per h
    const int lane = threadIdx.x & 31;
    const v4f* arow = (const v4f*)(attn + (size_t)b * SS);
    const v4f* erow = (const v4f*)(enc_tot + ((size_t)b * HH + h) * SS);
    float acc = 0.f;
    for (int i = lane; i < SS / 4; i += 32) {
        v4f a4 = arow[i];
        v4f e4 = __builtin_nontemporal_load(&erow[i]);
        acc += a4.x * e4.x + a4.y * e4.y + a4.z * e4.z + a4.w * e4.w;
    }
    for (int o = 16; o > 0; o >>= 1) acc += __shfl_down(acc, o, 32);
    if (lane == 0) {
        float hv = glu[(size_t)b * HH + h] + acc;
        hout[(size_t)b * HH + h]   = hv;
        x2next[(size_t)b * HH + h] = hv + pe_val(h, 2);
    }
}

// out1[b,h,0] = softmax over h of h3[b,:]
__global__ void out_softmax_h_kernel(const float* __restrict__ h3,
                                     float* __restrict__ out) {
    const int b = blockIdx.x;
    const int t = threadIdx.x;                  // 256
    __shared__ float red[256];
    const float* row = h3 + (size_t)b * HH;
    float m = -INFINITY;
    for (int i = t; i < HH; i += 256) m = fmaxf(m, row[i]);
    red[t] = m; __syncthreads();
    for (int o = 128; o > 0; o >>= 1) { if (t < o) red[t] = fmaxf(red[t], red[t + o]); __syncthreads(); }
    m = red[0]; __syncthreads();
    float s = 0.f;
    for (int i = t; i < HH; i += 256) { float e = __expf(row[i] - m); out[(size_t)b * HH + i] = e; s += e; }
    red[t] = s; __syncthreads();
    for (int o = 128; o > 0; o >>= 1) { if (t < o) red[t] += red[t + o]; __syncthreads(); }
    float inv = 1.0f / red[0];
    for (int i = t; i < HH; i += 256) out[(size_t)b * HH + i] *= inv;
}

// logits[b,v] = h3[b,:] . lin2_w[v,:] + lin2_b[v]; out2 = log_softmax over v
__global__ void out_logits_kernel(const float* __restrict__ h3,
                                  const float* __restrict__ w,
                                  const float* __restrict__ bias,
                                  float* __restrict__ out2) {
    const int b = blockIdx.x;
    const int t = threadIdx.x;                  // 128
    __shared__ float sh[HH];
    __shared__ float red[128];
    for (int i = t; i < HH; i += 128) sh[i] = h3[(size_t)b * HH + i];
    __syncthreads();
    float logit = -INFINITY;
    if (t < VV) {
        float acc = bias[t];
        const float* wr = w + (size_t)t * HH;
        for (int e = 0; e < HH; ++e) acc += sh[e] * wr[e];
        logit = acc;
    }
    red[t] = logit; __syncthreads();
    for (int o = 64; o > 0; o >>= 1) { if (t < o) red[t] = fmaxf(red[t], red[t + o]); __syncthreads(); }
    float m = red[0]; __syncthreads();
    red[t] = (t < VV) ? __expf(logit - m) : 0.0f; __syncthreads();
    for (int o = 64; o > 0; o >>= 1) { if (t < o) red[t] += red[t + o]; __syncthreads(); }
    float lse = m + __logf(red[0]);
    if (t < VV) out2[(size_t)b * VV + t] = logit - lse;
}

extern "C" void kernel_launch(void* const* d_in, const int* in_sizes, int n_in,
                              void* d_out, int out_size, void* d_ws, size_t ws_size,
                              hipStream_t stream) {
    const float* enc_out = (const float*)d_in[0];   // [B,H,S]
    const float* enc_tot = (const float*)d_in[1];   // [B,H,S]
    const int*   dec_in  = (const int*)d_in[2];     // [B]
    // d_in[3] = count (unused, first-step path)
    const float* emb     = (const float*)d_in[4];   // [V,E]
    const float* conv_w  = (const float*)d_in[5];   // [2H,E,K]
    const float* conv_b  = (const float*)d_in[6];   // [2H]
    const float* lin2_w  = (const float*)d_in[7];   // [V,H]
    const float* lin2_b  = (const float*)d_in[8];   // [V]

    float* ws = (float*)d_ws;
    float* const_y = ws;                        // 1024
    float* x2      = const_y + 1024;            // B*E   = 65536
    float* ybuf    = x2 + BB * EE;              // B*2H  = 131072
    float* glu     = ybuf + BB * 2 * HH;        // B*H   = 65536
    float* dvec    = glu + BB * HH;             // B*H   = 65536
    float* attn    = dvec + BB * HH;            // B*S   = 262144
    float* hbuf    = attn + BB * SS;            // B*H   = 65536

    float* out1 = (float*)d_out;                // [B,H,1] softmax(h3)
    float* out2 = out1 + BB * HH;               // [B,1,V] log_softmax(logits)

    const_y_kernel<<<4, 256, 0, stream>>>(emb, conv_w, const_y);
    init_x2_kernel<<<(BB * EE) / 256, 256, 0, stream>>>(emb, dec_in, x2);

    for (int l = 0; l < 3; ++l) {
        conv_wmma_kernel<<<dim3(BB / 16, (2 * HH) / 16), 32, 0, stream>>>(
            x2, conv_w, const_y, conv_b, ybuf);
        glu_d_kernel<<<(BB * HH) / 256, 256, 0, stream>>>(ybuf, x2, glu, dvec);
        scores_kernel<<<dim3(BB, SS / 512), 128, 0, stream>>>(dvec, enc_out, attn);
        softmax_kernel<<<BB, 256, 0, stream>>>(attn);
        context_kernel<<<dim3(BB, HH / 8), 256, 0, stream>>>(
            attn, enc_tot, glu, hbuf, x2);  // x2 becomes next layer's last column
    }

    out_softmax_h_kernel<<<BB, 256, 0, stream>>>(hbuf, out1);
    out_logits_kernel<<<BB, 128, 0, stream>>>(hbuf, lin2_w, lin2_b, out2);
}